// TSPEdgeEmbedding_2250562863229
// MI455X (gfx1250) — compile-verified
//
#include <hip/hip_runtime.h>

typedef __attribute__((ext_vector_type(2))) float v2f;
typedef __attribute__((ext_vector_type(8))) float v8f;

#define Bb    16
#define Nn    1000
#define Dd    128
#define Kk    50
#define NEDGE (Bb * Nn * Kk)            /* 800000  */
#define OFF_EI  (Bb * Nn * Dd)          /* 2048000 : start of edge_index */
#define OFF_EMB (OFF_EI + 2 * NEDGE)    /* 3648000 : start of edge_emb   */

// ---------------------------------------------------------------------------
// Kernel 1: per-node top-50 nearest neighbors.
// One block = 128 nodes of one batch. locs for the batch staged in LDS (8KB,
// broadcast reads). Each thread keeps an insertion-sorted top-50 (dist,idx)
// in LDS with SoA layout [j*BLK + tid] -> conflict-free across a wave.
// Writes edge_index (src/dst as float values) directly to d_out and the
// sorted distances to d_ws for the WMMA embedding kernel.
// ---------------------------------------------------------------------------
#define TPB1 128
__global__ __launch_bounds__(TPB1) void topk_kernel(const float* __restrict__ locs,
                                                    float* __restrict__ out,
                                                    float* __restrict__ ws_dist) {
    __shared__ float sx[Nn];
    __shared__ float sy[Nn];
    __shared__ float bd[Kk * TPB1];
    __shared__ int   bi[Kk * TPB1];

    const int t  = threadIdx.x;
    const int bq = blockIdx.y;
    const float* L = locs + bq * Nn * 2;
    for (int j = t; j < Nn; j += TPB1) { sx[j] = L[2 * j]; sy[j] = L[2 * j + 1]; }
    __syncthreads();

    const int i = blockIdx.x * TPB1 + t;
    if (i < Nn) {
        const float xi = sx[i], yi = sy[i];
        for (int j = 0; j < Kk; ++j) { bd[j * TPB1 + t] = 3.0e38f; bi[j * TPB1 + t] = 0; }

        for (int j = 0; j < Nn; ++j) {
            if (j == i) continue;                    // self-loop masked with +inf in ref
            const float dx = xi - sx[j];
            const float dy = yi - sy[j];
            const float d  = sqrtf(dx * dx + dy * dy + 1e-12f);
            if (d < bd[(Kk - 1) * TPB1 + t]) {       // beats current worst -> insert
                int pos = Kk - 1;
                while (pos > 0 && bd[(pos - 1) * TPB1 + t] > d) {
                    bd[pos * TPB1 + t] = bd[(pos - 1) * TPB1 + t];
                    bi[pos * TPB1 + t] = bi[(pos - 1) * TPB1 + t];
                    --pos;
                }
                bd[pos * TPB1 + t] = d;
                bi[pos * TPB1 + t] = j;              // strict '>' keeps tie order = index asc
            }
        }

        const int g  = bq * Nn + i;                  // global node id (src with offset)
        const int e0 = g * Kk;
        for (int j = 0; j < Kk; ++j) {
            out[OFF_EI + e0 + j]         = (float)g;
            out[OFF_EI + NEDGE + e0 + j] = (float)(bq * Nn + bi[j * TPB1 + t]);
            ws_dist[e0 + j]              = bd[j * TPB1 + t];
        }
    }
}

// ---------------------------------------------------------------------------
// Kernel 2: x = init_embeddings.reshape(B*n, D) -- straight b128 copy.
// ---------------------------------------------------------------------------
__global__ __launch_bounds__(256) void copy_x_kernel(const float4* __restrict__ src,
                                                     float4* __restrict__ dst) {
    const int idx = blockIdx.x * 256 + threadIdx.x;
    dst[idx] = src[idx];
}

// ---------------------------------------------------------------------------
// Kernel 3: edge_emb[e, :] = dist[e] * W[:] + b[:]  via V_WMMA_F32_16X16X4_F32.
// Rank-1 GEMM: A(16x4) has K=0 column = 16 distances, K=1..3 = 0;
// B(4x16) has K=0 row = 16 W values, rest 0; C preloaded with bias broadcast.
// One wave = one 16-edge M-tile x 8 N-chunks (D=128). No divergence -> EXEC
// stays all-ones around WMMA (ISA requirement).
// ---------------------------------------------------------------------------
#define TPB3 256
__global__ __launch_bounds__(TPB3) void emb_wmma_kernel(const float* __restrict__ dist,
                                                        const float* __restrict__ W,
                                                        const float* __restrict__ bias,
                                                        float* __restrict__ out) {
    const int lane = threadIdx.x & 31;
    const int wave = threadIdx.x >> 5;
    const int tile = blockIdx.x * (TPB3 / 32) + wave;   // 0..49999, exact
    const int e0   = tile * 16;
    const int l15  = lane & 15;
    const bool lo  = (lane < 16);

    // A-matrix (f32 16x4): VGPR0 lanes0-15 = K=0 (M=lane); everything else 0.
    float dv = dist[e0 + l15];                // all lanes load (clamped idx), select below
    dv = lo ? dv : 0.0f;
    v2f a; a.x = dv; a.y = 0.0f;

    const int mbase = (lane >> 4) << 3;       // 0 for lanes 0-15, 8 for lanes 16-31

#pragma unroll
    for (int c = 0; c < 8; ++c) {             // 8 chunks of 16 output channels
        float wv = W[c * 16 + l15];
        wv = lo ? wv : 0.0f;
        v2f bm; bm.x = wv; bm.y = 0.0f;       // B-matrix: K=0 row only

        const float bv = bias[c * 16 + l15];  // C[m][n] = b[n] for every row m
        v8f cc;
#pragma unroll
        for (int r = 0; r < 8; ++r) cc[r] = bv;

        // 8 args: (neg_a, A, neg_b, B, c_mod, C, reuse_a, reuse_b)
        v8f d = __builtin_amdgcn_wmma_f32_16x16x4_f32(
            false, a, false, bm, (short)0, cc, false, false);

        // D layout: VGPR r, lanes0-15 -> row r, lanes16-31 -> row r+8; col = lane%16.
#pragma unroll
        for (int r = 0; r < 8; ++r) {
            const int m = r + mbase;
            out[OFF_EMB + (e0 + m) * Dd + c * 16 + l15] = d[r];
        }
    }
}

extern "C" void kernel_launch(void* const* d_in, const int* in_sizes, int n_in,
                              void* d_out, int out_size, void* d_ws, size_t ws_size,
                              hipStream_t stream) {
    (void)in_sizes; (void)n_in; (void)out_size; (void)ws_size;
    const float* locs     = (const float*)d_in[0];   // [16,1000,2]
    const float* init_emb = (const float*)d_in[1];   // [16,1000,128]
    const float* W        = (const float*)d_in[2];   // [1,128]
    const float* bias     = (const float*)d_in[3];   // [128]
    float* out     = (float*)d_out;
    float* ws_dist = (float*)d_ws;                   // NEDGE floats (3.2 MB)

    // 1) top-k neighbors + edge_index + distances
    dim3 g1((Nn + TPB1 - 1) / TPB1, Bb);
    topk_kernel<<<g1, TPB1, 0, stream>>>(locs, out, ws_dist);

    // 2) x copy (2,048,000 floats = 512,000 float4)
    copy_x_kernel<<<(Bb * Nn * Dd / 4) / 256, 256, 0, stream>>>(
        (const float4*)init_emb, (float4*)out);

    // 3) edge embeddings via WMMA: 50,000 tiles / 8 waves per block = 6250 blocks
    emb_wmma_kernel<<<(NEDGE / 16) / (TPB3 / 32), TPB3, 0, stream>>>(
        ws_dist, W, bias, out + 0);
}